// Sinkhorn_14654428414008
// MI455X (gfx1250) — compile-verified
//
#include <hip/hip_runtime.h>

// Sinkhorn forward for MI455X (gfx1250).
// Strategy: never materialize the 268MB cost matrix. Recompute x.y^T tiles with
// V_WMMA_F32_16X16X4_F32 (K=D=3 padded to 4) and do an online logsumexp over
// columns directly in the WMMA accumulator layout. 41 softmin passes total.

#define NB     4
#define P      4096
#define CHUNK  1024
#define WAVES  8

typedef __attribute__((ext_vector_type(2))) float v2f;
typedef __attribute__((ext_vector_type(8))) float v8f;

__device__ __forceinline__ float fast_exp2(float x) { return __builtin_amdgcn_exp2f(x); }
__device__ __forceinline__ float fast_log2(float x) { return __builtin_amdgcn_logf(x); }

// blur=0.05, p=2 -> eps = 0.0025
#define C_EPS      0.0025f
#define C_INV_EPS  400.0f
#define C_LOG_B    (-8.317766166719343f)   /* -ln(4096) */
#define C_LOG2E    1.4426950408889634f
#define C_EPS_LN2  0.0017328679513998632f  /* eps * ln2 */
#define C_SCALE    577.0780163555854f      /* inv_eps * log2(e) */

// Computes, for 16 rows i of X:  fout_i = 0.5|x_i|^2 - eps*LSE_j( gt_j + (x_i.y_j)/eps )
// where gt_j = log_b + (g_j - 0.5|y_j|^2)/eps.  Symmetric in both Sinkhorn directions.
__global__ __launch_bounds__(256) void softmin_pass(
    const float* __restrict__ Xrow,   // [NB][P][3]  (rows of the pass)
    const float* __restrict__ Ycol,   // [NB][P][3]  (columns of the pass)
    const float* __restrict__ gin,    // [NB][P]     dual potential of columns
    float* __restrict__ fout)         // [NB][P]     output potential of rows
{
    __shared__ float sy0[CHUNK], sy1[CHUNK], sy2[CHUNK], sgt[CHUNK];
    __shared__ float lds_m[16][WAVES], lds_s[16][WAVES];

    const int n    = blockIdx.y;
    const int i0   = blockIdx.x * 16;
    const int t    = threadIdx.x;
    const int lane = t & 31;
    const int wave = t >> 5;

    const float* X = Xrow + (size_t)n * P * 3;
    const float* Y = Ycol + (size_t)n * P * 3;
    const float* G = gin  + (size_t)n * P;

    // A fragment: 16x4 f32, lanes 0-15 hold K=0,1 of row M=lane; lanes 16-31 hold K=2,3.
    const int r = i0 + (lane & 15);
    v2f a;
    if (lane < 16) { a[0] = X[r * 3 + 0]; a[1] = X[r * 3 + 1]; }
    else           { a[0] = X[r * 3 + 2]; a[1] = 0.0f;        }

    // Online LSE state: each lane owns 8 rows x 1 column-slot (WMMA C/D layout).
    float m[8], ss[8];
    #pragma unroll
    for (int e = 0; e < 8; ++e) { m[e] = -__builtin_inff(); ss[e] = 0.0f; }

    for (int jc = 0; jc < P; jc += CHUNK) {
        // Stage a 1024-column chunk of Y (+ column bias gt) into LDS.
        #pragma unroll
        for (int rr = 0; rr < CHUNK / 256; ++rr) {
            const int jl = t + rr * 256;
            const int j  = jc + jl;
            const float y0 = Y[j * 3 + 0];
            const float y1 = Y[j * 3 + 1];
            const float y2 = Y[j * 3 + 2];
            sy0[jl] = y0; sy1[jl] = y1; sy2[jl] = y2;
            const float ysq = y0 * y0 + y1 * y1 + y2 * y2;
            sgt[jl] = (C_LOG_B + (G[j] - 0.5f * ysq) * C_INV_EPS) * C_LOG2E;
        }
        __syncthreads();

        // 64 column-tiles per chunk, 8 per wave.
        #pragma unroll
        for (int k = 0; k < (CHUNK / 16) / WAVES; ++k) {
            const int jj = (k * WAVES + wave) * 16 + (lane & 15);
            // B fragment: 4x16 f32, lanes 0-15 hold K=0,1 of col N=lane; lanes 16-31 K=2,3.
            const float* plo = (lane < 16) ? sy0 : sy2;
            v2f b;
            b[0] = plo[jj];
            b[1] = (lane < 16) ? sy1[jj] : 0.0f;
            const float vbase = sgt[jj];

            v8f c = {0.f, 0.f, 0.f, 0.f, 0.f, 0.f, 0.f, 0.f};
            c = __builtin_amdgcn_wmma_f32_16x16x4_f32(
                    false, a, false, b, (short)0, c, false, false);

            // Branchless online logsumexp (base 2): one v_exp_f32 per element.
            #pragma unroll
            for (int e = 0; e < 8; ++e) {
                const float v   = fmaf(c[e], C_SCALE, vbase);
                const float d   = v - m[e];
                const float ex  = fast_exp2(-__builtin_fabsf(d));
                const float shi = fmaf(ss[e], ex, 1.0f);  // new max
                const float slo = ss[e] + ex;             // old max keeps
                const bool  up  = d > 0.0f;
                ss[e] = up ? shi : slo;
                m[e]  = up ? v   : m[e];
            }
        }
        __syncthreads();
    }

    // Merge the 16 column-slots (lanes) that share each row.
    #pragma unroll
    for (int off = 8; off >= 1; off >>= 1) {
        #pragma unroll
        for (int e = 0; e < 8; ++e) {
            const float m2 = __shfl_xor(m[e],  off, 32);
            const float s2 = __shfl_xor(ss[e], off, 32);
            const float M  = fmaxf(m[e], m2);
            ss[e] = fmaf(ss[e], fast_exp2(m[e] - M), s2 * fast_exp2(m2 - M));
            m[e]  = M;
        }
    }

    // One writer per 16-lane group: lanes 0 (rows 0-7) and 16 (rows 8-15).
    if ((lane & 15) == 0) {
        const int rbase = (lane >> 4) * 8;
        #pragma unroll
        for (int e = 0; e < 8; ++e) {
            lds_m[rbase + e][wave] = m[e];
            lds_s[rbase + e][wave] = ss[e];
        }
    }
    __syncthreads();

    // Merge the 8 waves and finalize 16 rows.
    if (t < 16) {
        float M = lds_m[t][0], S = lds_s[t][0];
        #pragma unroll
        for (int w = 1; w < WAVES; ++w) {
            const float m2 = lds_m[t][w], s2 = lds_s[t][w];
            const float Mx = fmaxf(M, m2);
            S = fmaf(S, fast_exp2(M - Mx), s2 * fast_exp2(m2 - Mx));
            M = Mx;
        }
        const int row = i0 + t;
        const float x0 = X[row * 3 + 0];
        const float x1 = X[row * 3 + 1];
        const float x2 = X[row * 3 + 2];
        const float rsq = x0 * x0 + x1 * x1 + x2 * x2;
        fout[(size_t)n * P + row] = 0.5f * rsq - C_EPS_LN2 * (M + fast_log2(S));
    }
}

__global__ __launch_bounds__(256) void final_reduce(
    const float* __restrict__ f, const float* __restrict__ g,
    float* __restrict__ out)
{
    __shared__ float acc[256];
    float s = 0.0f;
    for (int i = threadIdx.x; i < NB * P; i += 256) s += f[i] + g[i];
    acc[threadIdx.x] = s;
    __syncthreads();
    #pragma unroll
    for (int o = 128; o > 0; o >>= 1) {
        if (threadIdx.x < o) acc[threadIdx.x] += acc[threadIdx.x + o];
        __syncthreads();
    }
    if (threadIdx.x == 0) out[0] = acc[0] * (1.0f / (float)P);
}

extern "C" void kernel_launch(void* const* d_in, const int* in_sizes, int n_in,
                              void* d_out, int out_size, void* d_ws, size_t ws_size,
                              hipStream_t stream) {
    const float* x = (const float*)d_in[0];   // [4,4096,3]
    const float* y = (const float*)d_in[1];   // [4,4096,3]
    float* f = (float*)d_ws;                  // [4,4096]
    float* g = f + (size_t)NB * P;            // [4,4096]

    hipMemsetAsync(g, 0, (size_t)NB * P * sizeof(float), stream);

    const dim3 grid(P / 16, NB);
    const dim3 block(256);

    for (int it = 0; it < 20; ++it) {
        softmin_pass<<<grid, block, 0, stream>>>(x, y, g, f);  // f = softmin_j(g)
        softmin_pass<<<grid, block, 0, stream>>>(y, x, f, g);  // g = softmin_i(f)
    }
    // Final extrapolation: f_fin = softmin_j(g20); g_fin = softmin_i(f20) == g20 already.
    softmin_pass<<<grid, block, 0, stream>>>(x, y, g, f);

    final_reduce<<<1, block, 0, stream>>>(f, g, (float*)d_out);
}